// SimplePPMIEncoder_28948079575219
// MI455X (gfx1250) — compile-verified
//
#include <hip/hip_runtime.h>

typedef float v2f __attribute__((ext_vector_type(2)));
typedef float v8f __attribute__((ext_vector_type(8)));

#define N_NODES 100000
#define N_EDGES 1600000
#define IN_DIM  128
#define HID_DIM 64
#define OUT_DIM 64

// ---------------------------------------------------------------- utilities

__global__ void zero_f32(float* __restrict__ p, int n) {
  int i = blockIdx.x * blockDim.x + threadIdx.x;
  if (i < n) p[i] = 0.0f;
}

__global__ void deg_accum(const int* __restrict__ row,
                          const float* __restrict__ w,
                          float* __restrict__ deg) {
  int e = blockIdx.x * blockDim.x + threadIdx.x;
  if (e < N_EDGES) unsafeAtomicAdd(&deg[row[e]], w[e]);
}

__global__ void norm_compute(const int* __restrict__ row,
                             const int* __restrict__ col,
                             const float* __restrict__ w,
                             const float* __restrict__ deg,
                             float* __restrict__ norm) {
  int e = blockIdx.x * blockDim.x + threadIdx.x;
  if (e >= N_EDGES) return;
  float dr = deg[row[e]];
  float dc = deg[col[e]];
  float ir = dr > 0.0f ? __frsqrt_rn(fmaxf(dr, 1e-12f)) : 0.0f;
  float ic = dc > 0.0f ? __frsqrt_rn(fmaxf(dc, 1e-12f)) : 0.0f;
  norm[e] = ir * w[e] * ic;
}

// out[n][d] = bias[d]  (bias folded into the scatter-add accumulator init)
__global__ void init_bias(float* __restrict__ out, const float* __restrict__ b) {
  int i = blockIdx.x * blockDim.x + threadIdx.x;
  if (i < N_NODES * 64) out[i] = b[i & 63];
}

// ---------------------------------------------------------------- WMMA GEMM
// C[M x 64] = op(A)[M x K_DIM] @ B[K_DIM x 64], op = identity or ReLU.
// One wave32 per 16x16 C tile, V_WMMA_F32_16X16X4_F32 accumulation.
// A-frag (16x4 f32, ISA 7.12.2): lanes 0-15 hold rows M=0..15 with K=0,1 in
// v[0],v[1]; lanes 16-31 hold K=2,3.  B-frag (4x16): lanes hold column N,
// same K split across lane halves.  C/D: VGPR r -> M=r (lanes 0-15) or
// M=r+8 (lanes 16-31), N = lane&15.
template <int K_DIM, bool RELU_A>
__global__ void gemm_wmma(const float* __restrict__ A,
                          const float* __restrict__ B,
                          float* __restrict__ C) {
  const int wave  = threadIdx.x >> 5;
  const int gw    = blockIdx.x * 8 + wave;   // 8 waves / block
  const int mTile = gw >> 2;                 // N = 64 -> 4 n-tiles
  const int nTile = gw & 3;
  const int lane  = threadIdx.x & 31;
  const int hlf   = lane >> 4;               // lane half selects K pair
  const int l     = lane & 15;
  const int m0    = mTile * 16;
  const int n0    = nTile * 16;

  v8f acc = {};
  const float* aRow = A + (size_t)(m0 + l) * K_DIM + 2 * hlf;
  const float* bCol = B + (size_t)(2 * hlf) * 64 + n0 + l;

#pragma unroll
  for (int k = 0; k < K_DIM; k += 4) {
    v2f a, b;
    a.x = aRow[k + 0];
    a.y = aRow[k + 1];
    if (RELU_A) { a.x = fmaxf(a.x, 0.0f); a.y = fmaxf(a.y, 0.0f); }
    b.x = bCol[(size_t)k * 64];
    b.y = bCol[(size_t)k * 64 + 64];
    // 8 args: (neg_a, A, neg_b, B, c_mod, C, reuse_a, reuse_b)
    acc = __builtin_amdgcn_wmma_f32_16x16x4_f32(false, a, false, b,
                                                (short)0, acc, false, false);
  }

  float* cOut = C + (size_t)(m0 + 8 * hlf) * 64 + n0 + l;
#pragma unroll
  for (int r = 0; r < 8; ++r) cOut[(size_t)r * 64] = acc[r];
}

// ---------------------------------------------------------------- scatter
// agg[col[e]][:] += h[row[e]][:] * norm[e];  16 threads per edge, float4 each.
__global__ void aggregate(const float* __restrict__ h,
                          const float* __restrict__ norm,
                          const int* __restrict__ row,
                          const int* __restrict__ col,
                          float* __restrict__ agg) {
  int tid = blockIdx.x * blockDim.x + threadIdx.x;
  int e   = tid >> 4;
  int sub = tid & 15;
  if (e >= N_EDGES) return;
  int   r  = row[e];
  int   c  = col[e];
  float nw = norm[e];
  const float4 v = *(const float4*)(h + (size_t)r * 64 + sub * 4);
  float* dst = agg + (size_t)c * 64 + sub * 4;
  unsafeAtomicAdd(dst + 0, v.x * nw);
  unsafeAtomicAdd(dst + 1, v.y * nw);
  unsafeAtomicAdd(dst + 2, v.z * nw);
  unsafeAtomicAdd(dst + 3, v.w * nw);
}

// ---------------------------------------------------------------- launcher

extern "C" void kernel_launch(void* const* d_in, const int* in_sizes, int n_in,
                              void* d_out, int out_size, void* d_ws, size_t ws_size,
                              hipStream_t stream) {
  const float* x  = (const float*)d_in[0];
  const int*   ei = (const int*)  d_in[1];   // [2, N_EDGES]
  const float* ew = (const float*)d_in[2];
  const float* W1 = (const float*)d_in[3];
  const float* b1 = (const float*)d_in[4];
  const float* W2 = (const float*)d_in[5];
  const float* b2 = (const float*)d_in[6];
  const int* row = ei;
  const int* col = ei + N_EDGES;

  float* ws   = (float*)d_ws;
  float* deg  = ws;               // 100,000 f32
  float* norm = ws + 100000;      // 1,600,000 f32
  float* h    = ws + 1700000;     // 6,400,000 f32
  float* agg1 = ws + 8100000;     // 6,400,000 f32
  float* h2   = h;                // h dead after agg1; reuse for layer-2 GEMM
  float* out  = (float*)d_out;

  // normalization
  zero_f32   <<<(N_NODES + 255) / 256, 256, 0, stream>>>(deg, N_NODES);
  deg_accum  <<<(N_EDGES + 255) / 256, 256, 0, stream>>>(row, ew, deg);
  norm_compute<<<(N_EDGES + 255) / 256, 256, 0, stream>>>(row, col, ew, deg, norm);

  // layer 1: h = x @ W1 ; agg1 = scatter(h*norm) + b1
  gemm_wmma<IN_DIM, false><<<3125, 256, 0, stream>>>(x, W1, h);     // 6250*4 tiles / 8 waves
  init_bias  <<<(N_NODES * 64 + 255) / 256, 256, 0, stream>>>(agg1, b1);
  aggregate  <<<N_EDGES * 16 / 256, 256, 0, stream>>>(h, norm, row, col, agg1);

  // layer 2: h2 = relu(agg1) @ W2 ; out = scatter(h2*norm) + b2
  gemm_wmma<HID_DIM, true><<<3125, 256, 0, stream>>>(agg1, W2, h2);
  init_bias  <<<(N_NODES * 64 + 255) / 256, 256, 0, stream>>>(out, b2);
  aggregate  <<<N_EDGES * 16 / 256, 256, 0, stream>>>(h2, norm, row, col, out);
}